// Transformer_46067819217535
// MI455X (gfx1250) — compile-verified
//
#include <hip/hip_runtime.h>
#include <math.h>

typedef float v2f __attribute__((ext_vector_type(2)));
typedef float v8f __attribute__((ext_vector_type(8)));

#define B_    32
#define CIN   3
#define HIN   224
#define WIN   224
#define EMB   192
#define OH    112
#define OW    112
#define PH_   56
#define PW_   56
#define CB    12          // channel blocks of 16 (192/16)
#define KK    27          // reduction dim of implicit GEMM (3*3*3)
#define LDSW  226         // padded input row (cols -1..224)
#define NEGBIG (-3.402823466e+38f)

__device__ __forceinline__ float gelu_erf(float z) {
    return 0.5f * z * (1.0f + erff(z * 0.70710678118654752f));
}

// k -> (c, kh, kw) decomposition, k = c*9 + kh*3 + kw
__device__ __forceinline__ void kdec(int k, int& c, int& kh, int& kw) {
    c = k / 9;
    int r = k - 9 * c;
    kh = r / 3;
    kw = r - 3 * kh;
}

// CDNA5 async global->LDS copy (4B per lane), tracked by ASYNCcnt.
// LDS offset = low 32 bits of the flat shared pointer (aperture truncation).
__device__ __forceinline__ void async_g2l_b32(const float* gptr, float* lptr) {
    unsigned lds_off = (unsigned)(unsigned long long)lptr;
    unsigned long long ga = (unsigned long long)gptr;
    asm volatile("global_load_async_to_lds_b32 %0, %1, off"
                 :: "v"(lds_off), "v"(ga) : "memory");
}
__device__ __forceinline__ void wait_async0() {
    asm volatile("s_wait_asynccnt 0x0" ::: "memory");
}

// ---------------------------------------------------------------------------
// Pass 1: per-(batch, channel-block) conv statistics (sum, sumsq) via WMMA.
// Grid: B_*CB blocks, 128 threads (4 waves).
// ---------------------------------------------------------------------------
__global__ __launch_bounds__(128) void conv_stats_kernel(
    const float* __restrict__ x, const float* __restrict__ w,
    const float* __restrict__ bias, float* __restrict__ partial)
{
    __shared__ float lx[CIN * 3 * LDSW];   // 3 input rows x 3 ch, padded
    __shared__ float red[2][128];

    const int blk  = blockIdx.x;           // b*CB + cb
    const int b    = blk / CB;
    const int cb   = blk - b * CB;
    const int e0   = cb * 16;
    const int tid  = threadIdx.x;
    const int wave = tid >> 5;
    const int lane = tid & 31;
    const int half = lane >> 4;
    const int lm   = lane & 15;
    const int e    = e0 + lm;

    // Preload B-matrix fragments (weights) once: 7 K-steps of K=4.
    v2f bf[7];
    #pragma unroll
    for (int j = 0; j < 7; ++j) {
        int k0 = 4 * j + 2 * half;                    // <= 26
        int k1 = k0 + 1;
        int k1c = (k1 < KK) ? k1 : 0;                 // clamp address
        float w1 = w[e * KK + k1c];
        bf[j].x = w[e * KK + k0];
        bf[j].y = (k1 < KK) ? w1 : 0.0f;
    }
    const float bia = bias[e];

    float s_sum = 0.0f, s_sq = 0.0f;

    for (int oh = 0; oh < OH; ++oh) {
        __syncthreads();
        // Stage input rows 2*oh-1 .. 2*oh+1, zero-padded, async into LDS
        for (int idx = tid; idx < CIN * 3 * LDSW; idx += 128) {
            int col = idx % LDSW;
            int t   = idx / LDSW;     // c*3 + ir
            int ir  = t % 3;
            int c   = t / 3;
            int gr  = 2 * oh - 1 + ir;
            int gc  = col - 1;
            if ((unsigned)gr < HIN && (unsigned)gc < WIN) {
                const float* gp = &x[((b * CIN + c) * HIN + gr) * WIN + gc];
                async_g2l_b32(gp, &lx[idx]);
                // sparse prefetch of the rows needed next oh iteration
                if (((idx & 15) == 0) && (gr + 2 < HIN))
                    __builtin_prefetch(gp + 2 * WIN, 0, 1);
            } else {
                lx[idx] = 0.0f;
            }
        }
        wait_async0();
        __syncthreads();

        for (int t = wave; t < 7; t += 4) {          // 7 M-tiles of 16 cols
            const int ow = t * 16 + lm;              // A row = lane&15
            v8f acc = {};
            #pragma unroll
            for (int j = 0; j < 7; ++j) {
                int k0 = 4 * j + 2 * half;
                int k1 = k0 + 1;
                int k1c = (k1 < KK) ? k1 : k0;       // clamped (valid) index
                int c0, kh0, kw0, c1, kh1, kw1;
                kdec(k0, c0, kh0, kw0);
                kdec(k1c, c1, kh1, kw1);
                float a0 = lx[(c0 * 3 + kh0) * LDSW + 2 * ow + kw0];
                float a1 = lx[(c1 * 3 + kh1) * LDSW + 2 * ow + kw1];
                v2f a;
                a.x = a0;
                a.y = (k1 < KK) ? a1 : 0.0f;         // value select, no branch
                acc = __builtin_amdgcn_wmma_f32_16x16x4_f32(
                    false, a, false, bf[j], (short)0, acc, false, false);
            }
            #pragma unroll
            for (int i = 0; i < 8; ++i) {
                float y = acc[i] + bia;              // all 8 rows: channel e
                s_sum += y;
                s_sq  += y * y;
            }
        }
    }

    // Deterministic block reduction: 8 contributors per channel, fixed order.
    __syncthreads();
    red[0][tid] = s_sum;
    red[1][tid] = s_sq;
    __syncthreads();
    if (tid < 16) {
        float a = 0.0f, q = 0.0f;
        for (int i = tid; i < 128; i += 16) { a += red[0][i]; q += red[1][i]; }
        partial[blk * 16 + tid] = a;
        partial[B_ * CB * 16 + blk * 16 + tid] = q;
    }
}

// ---------------------------------------------------------------------------
// Pass 1b: reduce partials -> per-channel BN scale/shift. 1 block, 192 thr.
// ---------------------------------------------------------------------------
__global__ void stats_reduce_kernel(
    const float* __restrict__ partial, const float* __restrict__ gamma,
    const float* __restrict__ beta, float* __restrict__ scaleshift)
{
    int e = threadIdx.x;
    if (e < EMB) {
        int cb = e >> 4, c16 = e & 15;
        float s = 0.0f, q = 0.0f;
        for (int b = 0; b < B_; ++b) {               // fixed order
            int blk = b * CB + cb;
            s += partial[blk * 16 + c16];
            q += partial[B_ * CB * 16 + blk * 16 + c16];
        }
        const float N = (float)(B_ * OH * OW);
        float mean = s / N;
        float var  = q / N - mean * mean;
        float rstd = rsqrtf(var + 1e-5f);
        float g    = gamma[e] * rstd;
        scaleshift[e]       = g;
        scaleshift[EMB + e] = beta[e] - mean * g;
    }
}

// ---------------------------------------------------------------------------
// Pass 2: fused conv (WMMA) + bias + BN + GELU + 3x3/s2 maxpool.
// Grid: B_*PH_*CB blocks, 128 threads (4 waves).
// ---------------------------------------------------------------------------
__global__ __launch_bounds__(128) void fused_patch_embed_kernel(
    const float* __restrict__ x, const float* __restrict__ w,
    const float* __restrict__ bias, const float* __restrict__ scaleshift,
    float* __restrict__ out)
{
    __shared__ float lx[CIN * 7 * LDSW];   // 7 input rows x 3 ch (19.0 KB)
    __shared__ float lc[3 * 114 * 16];     // 3 conv rows x (cols -1..112) x 16ch

    const int blk  = blockIdx.x;           // ((b*PH)+ph)*CB + cb
    const int cb   = blk % CB;
    const int t2   = blk / CB;
    const int ph   = t2 % PH_;
    const int b    = t2 / PH_;
    const int e0   = cb * 16;
    const int tid  = threadIdx.x;
    const int wave = tid >> 5;
    const int lane = tid & 31;
    const int half = lane >> 4;
    const int lm   = lane & 15;
    const int e    = e0 + lm;

    v2f bf[7];
    #pragma unroll
    for (int j = 0; j < 7; ++j) {
        int k0 = 4 * j + 2 * half;
        int k1 = k0 + 1;
        int k1c = (k1 < KK) ? k1 : 0;
        float w1 = w[e * KK + k1c];
        bf[j].x = w[e * KK + k0];
        bf[j].y = (k1 < KK) ? w1 : 0.0f;
    }
    const float bia = bias[e];
    const float scl = scaleshift[e];
    const float sft = scaleshift[EMB + e];

    // Stage input rows 4*ph-3 .. 4*ph+3 (zero-padded), async into LDS
    for (int idx = tid; idx < CIN * 7 * LDSW; idx += 128) {
        int col = idx % LDSW;
        int t   = idx / LDSW;     // c*7 + ir
        int ir  = t % 7;
        int c   = t / 7;
        int gr  = 4 * ph - 3 + ir;
        int gc  = col - 1;
        if ((unsigned)gr < HIN && (unsigned)gc < WIN) {
            async_g2l_b32(&x[((b * CIN + c) * HIN + gr) * WIN + gc], &lx[idx]);
        } else {
            lx[idx] = 0.0f;
        }
    }
    // -inf pads at conv cols -1 and 112 (pool window edges)
    if (tid < 96) {
        int c16  = tid & 15;
        int z    = tid >> 4;          // 0..5
        int cr   = z >> 1;
        int side = z & 1;
        lc[(cr * 114 + (side ? 113 : 0)) * 16 + c16] = NEGBIG;
    }
    wait_async0();
    __syncthreads();

    // 3 conv rows x 112 cols = 21 M-tiles of 16
    for (int t = wave; t < 21; t += 4) {
        const int cr = t / 7;                 // conv row within the 3-row band
        const int tw = t - cr * 7;
        const int ow = tw * 16 + lm;
        const int oh = 2 * ph - 1 + cr;       // uniform per tile
        v8f acc = {};
        #pragma unroll
        for (int j = 0; j < 7; ++j) {
            int k0 = 4 * j + 2 * half;
            int k1 = k0 + 1;
            int k1c = (k1 < KK) ? k1 : k0;
            int c0, kh0, kw0, c1, kh1, kw1;
            kdec(k0, c0, kh0, kw0);
            kdec(k1c, c1, kh1, kw1);
            float a0 = lx[(c0 * 7 + (2 * cr + kh0)) * LDSW + 2 * ow + kw0];
            float a1 = lx[(c1 * 7 + (2 * cr + kh1)) * LDSW + 2 * ow + kw1];
            v2f a;
            a.x = a0;
            a.y = (k1 < KK) ? a1 : 0.0f;
            acc = __builtin_amdgcn_wmma_f32_16x16x4_f32(
                false, a, false, bf[j], (short)0, acc, false, false);
        }
        const bool valid = (oh >= 0);         // oh <= 111 always holds
        #pragma unroll
        for (int i = 0; i < 8; ++i) {
            int m   = i + 8 * half;           // D row
            int owi = tw * 16 + m;
            float y = acc[i] + bia;
            float zz = y * scl + sft;         // BN
            float g  = valid ? gelu_erf(zz) : NEGBIG;
            lc[(cr * 114 + owi + 1) * 16 + lm] = g;
        }
    }
    __syncthreads();

    // 3x3 stride-2 maxpool over the 3 conv rows -> one pooled row
    for (int o = tid; o < PW_ * 16; o += 128) {
        int c16 = o & 15;
        int pw  = o >> 4;
        float m = NEGBIG;
        #pragma unroll
        for (int cr = 0; cr < 3; ++cr)
            #pragma unroll
            for (int dc = 0; dc < 3; ++dc)
                m = fmaxf(m, lc[(cr * 114 + 2 * pw + dc) * 16 + c16]);
        out[((b * EMB + (e0 + c16)) * PH_ + ph) * PW_ + pw] = m;
    }
}

// ---------------------------------------------------------------------------
extern "C" void kernel_launch(void* const* d_in, const int* in_sizes, int n_in,
                              void* d_out, int out_size, void* d_ws, size_t ws_size,
                              hipStream_t stream)
{
    const float* x     = (const float*)d_in[0];
    const float* w     = (const float*)d_in[1];   // [192,3,3,3] = [192][27]
    const float* bias  = (const float*)d_in[2];
    const float* gamma = (const float*)d_in[3];
    const float* beta  = (const float*)d_in[4];
    float* out = (float*)d_out;

    float* partial    = (float*)d_ws;               // 2 * (B_*CB*16) floats
    float* scaleshift = partial + 2 * B_ * CB * 16; // 2 * EMB floats

    conv_stats_kernel<<<B_ * CB, 128, 0, stream>>>(x, w, bias, partial);
    stats_reduce_kernel<<<1, 192, 0, stream>>>(partial, gamma, beta, scaleshift);
    fused_patch_embed_kernel<<<B_ * PH_ * CB, 128, 0, stream>>>(
        x, w, bias, scaleshift, out);
}